// BaseIEModel_12635793784907
// MI455X (gfx1250) — compile-verified
//
#include <hip/hip_runtime.h>
#include <hip/hip_bf16.h>

// ---------------------------------------------------------------------------
// CDNA5 (gfx1250) implementation, fp32 WMMA (V_WMMA_F32_16X16X4_F32) based.
// Shapes: B=32, S=1024, D=1024, T=4, NER=17, RE=16, H=512, P=10, Sm=1023,
// Dx=1041.
//
// Fragment layouts are arranged so every WMMA operand is one aligned
// contiguous ds_load_b64:
//   A: row stride 34 floats (even) -> (k+2h, k+2h+1) pair is 8B aligned.
//   B: pair-interleaved float2 Bs2[kp][n] = (B[2kp][n], B[2kp+1][n]) with a
//      160-float row stride (160 mod 64 == 32) so lanes 0-15 (row kp) and
//      lanes 16-31 (row kp+1) use disjoint LDS bank halves.
// ---------------------------------------------------------------------------

typedef __attribute__((ext_vector_type(2))) float v2f;
typedef __attribute__((ext_vector_type(8))) float v8f;

#define BN_  32
#define S_   1024
#define D_   1024
#define T_   4
#define NER_ 17
#define RE_  16
#define H_   512
#define P_   10
#define SM_  1023
#define DX_  1041

__device__ __forceinline__ v8f wmma_f32(v2f a, v2f b, v8f c) {
  // 8 args: (neg_a, A, neg_b, B, c_mod, C, reuse_a, reuse_b)
  return __builtin_amdgcn_wmma_f32_16x16x4_f32(
      false, a, false, b, (short)0, c, false, false);
}

// ---------------------------------------------------------------------------
// K1: ner = relu(h @ W1 + b1) @ W2 + b2, fused. Block tile: 128 rows, loops
// N(=1024 of W1) in chunks of 64 with a K-staged WMMA GEMM; relu'd chunk is
// contracted with W2[nchunk, 0:17] into an LDS accumulator (deterministic,
// thread-owned slots). Writes ner to workspace and ner_o rows to d_out.
// ---------------------------------------------------------------------------
#define K1_BM 128
#define K1_BN 64
#define K1_KB 32

__global__ __launch_bounds__(256) void k1_ner(
    const float* __restrict__ h, const float* __restrict__ W1,
    const float* __restrict__ b1g, const float* __restrict__ W2,
    const float* __restrict__ b2g, float* __restrict__ ner_ws,
    float* __restrict__ out_ner)
{
  // LDS layout (floats):
  //   union region [0, 8320):
  //       As[128][34] (4352) + Bs2[16][160] (2560) = 6912   OR   Rs[128][65] (8320)
  //   nacc [8320, 10496): 128*17
  //   W2s  [10496, 11584): 64*17
  __shared__ __align__(16) float smem[11584];
  float* As   = smem;               // [128][34]
  float* Bs2  = smem + 128 * 34;    // [16][160] pair-interleaved
  float* Rs   = smem;               // [128][65], aliases staging region
  float* nacc = smem + 8320;        // [128][17]
  float* W2s  = nacc + 128 * NER_;  // [64][17]

  const int tid   = threadIdx.x;
  const int wave  = tid >> 5;
  const int lane  = tid & 31;
  const int lhalf = lane >> 4;
  const int l16   = lane & 15;
  const int m0    = blockIdx.x * K1_BM;

  for (int s = tid; s < K1_BM * NER_; s += 256) nacc[s] = 0.f;

  for (int n0 = 0; n0 < 1024; n0 += K1_BN) {
    __syncthreads();  // prev chunk's stage-B done before W2s overwrite
    for (int s = tid; s < K1_BN * NER_; s += 256) {
      int kk = s / NER_, o = s % NER_;
      W2s[s] = W2[(size_t)(n0 + kk) * NER_ + o];
    }
    v8f cacc[4];
    for (int t = 0; t < 4; ++t)
      for (int r = 0; r < 8; ++r) cacc[t][r] = 0.f;

    for (int k0 = 0; k0 < 1024; k0 += K1_KB) {
      __syncthreads();  // staging region free (Rs of prev chunk consumed)
      // As: 128x32 floats, float4 global loads, stride-34 LDS rows
      for (int it = 0; it < 4; ++it) {
        int fi = tid + 256 * it;                 // 1024 float4
        int row = fi >> 3, c4 = (fi & 7) << 2;
        float4 v = *(const float4*)(h + (size_t)(m0 + row) * D_ + k0 + c4);
        float* dst = As + row * 34 + c4;
        dst[0] = v.x; dst[1] = v.y; dst[2] = v.z; dst[3] = v.w;
      }
      // Bs2: 32x64 floats, pair-interleaved (kp = k>>1, parity = k&1)
      for (int it = 0; it < 2; ++it) {
        int fi = tid + 256 * it;                 // 512 float4
        int row = fi >> 4, c4 = (fi & 15) << 2;
        float4 v = *(const float4*)(W1 + (size_t)(k0 + row) * 1024 + n0 + c4);
        float* dst = Bs2 + (row >> 1) * 160 + (row & 1);
        dst[(c4 + 0) * 2] = v.x; dst[(c4 + 1) * 2] = v.y;
        dst[(c4 + 2) * 2] = v.z; dst[(c4 + 3) * 2] = v.w;
      }
      // prefetch next k-stage tiles (global_prefetch_b8)
      if (k0 + K1_KB < 1024) {
        int row = tid >> 1, halfsel = tid & 1;
        __builtin_prefetch(h + (size_t)(m0 + row) * D_ + (k0 + K1_KB) + halfsel * 16, 0, 0);
        if (tid < 64) {
          int r2 = tid >> 1, h2 = tid & 1;
          __builtin_prefetch(W1 + (size_t)(k0 + K1_KB + r2) * 1024 + n0 + h2 * 32, 0, 0);
        }
      }
      __syncthreads();
      const float* arow  = As + (16 * wave + l16) * 34 + 2 * lhalf;
      const float* bbase = Bs2 + lhalf * 160 + l16 * 2;
      for (int k = 0; k < K1_KB; k += 4) {
        v2f a = *(const v2f*)(arow + k);
        const float* bk = bbase + (k >> 1) * 160;
        v2f bt0 = *(const v2f*)(bk);
        v2f bt1 = *(const v2f*)(bk + 32);
        v2f bt2 = *(const v2f*)(bk + 64);
        v2f bt3 = *(const v2f*)(bk + 96);
        cacc[0] = wmma_f32(a, bt0, cacc[0]);
        cacc[1] = wmma_f32(a, bt1, cacc[1]);
        cacc[2] = wmma_f32(a, bt2, cacc[2]);
        cacc[3] = wmma_f32(a, bt3, cacc[3]);
      }
    }
    __syncthreads();  // all reads of staging region done before Rs (alias) write
    for (int nt = 0; nt < 4; ++nt) {
      int col = nt * 16 + l16;
      float bb = b1g[n0 + col];
      for (int r = 0; r < 8; ++r) {
        int row = 16 * wave + r + 8 * lhalf;
        float v = cacc[nt][r] + bb;
        Rs[row * 65 + col] = v > 0.f ? v : 0.f;
      }
    }
    __syncthreads();
    // stage B: nacc[m][o] += Rs[m][:] @ W2s[:][o]  (thread-owned slots)
    for (int s = tid; s < K1_BM * NER_; s += 256) {
      int m = s / NER_, o = s % NER_;
      float acc = nacc[s];
      const float* rr  = Rs + m * 65;
      const float* w2c = W2s + o;
      for (int kk = 0; kk < K1_BN; ++kk) acc += rr[kk] * w2c[kk * NER_];
      nacc[s] = acc;
    }
  }
  __syncthreads();
  for (int s = tid; s < K1_BM * NER_; s += 256) {
    int m = s / NER_, o = s % NER_;
    float val = nacc[s] + b2g[o];
    int gm = m0 + m;
    ner_ws[(size_t)gm * NER_ + o] = val;
    int bb = gm >> 10, ss = gm & 1023;
    if (ss >= 1)
      out_ner[((size_t)bb * SM_ + (ss - 1)) * NER_ + o] = val;
  }
}

// ---------------------------------------------------------------------------
// K2: per-batch argmax + span state machine + ordered top-P selection.
// Reproduces the reference exactly: suffix-cummin nextE, forward accB scan,
// then first P valid entries in (t, kind, j) lexicographic order.
// ---------------------------------------------------------------------------
__global__ __launch_bounds__(256) void k2_spans(
    const float* __restrict__ ner_ws, int* __restrict__ j0_ws,
    int* __restrict__ e_ws, int* __restrict__ valid_ws,
    int* __restrict__ pos_ws, float* __restrict__ out_pp)
{
  __shared__ int amax[SM_];
  __shared__ int nextE[T_][SM_];
  __shared__ unsigned char accB[T_][SM_];
  __shared__ int selpos[P_];

  int b = blockIdx.x, tid = threadIdx.x;
  for (int j = tid; j < SM_; j += 256) {
    const float* row = ner_ws + (size_t)(b * S_ + j + 1) * NER_;
    float best = row[0]; int bi = 0;
    for (int o = 1; o < NER_; ++o) {
      float v = row[o];
      if (v > best) { best = v; bi = o; }
    }
    amax[j] = bi;
  }
  __syncthreads();
  if (tid < T_) {
    int t = tid;
    int mn = SM_;
    for (int j = SM_ - 1; j >= 0; --j) {
      nextE[t][j] = mn;                               // suff[j+1]
      int ep = (amax[j] == 3 + 4 * t) ? j : SM_;
      if (ep < mn) mn = ep;
    }
    int end = -1;
    for (int j = 0; j < SM_; ++j) {
      int acc = (amax[j] == 1 + 4 * t) && (j > end);
      accB[t][j] = (unsigned char)acc;
      if (acc) end = nextE[t][j];
    }
  }
  __syncthreads();
  if (tid == 0) {
    int cnt = 0;
    for (int t = 0; t < T_ && cnt < P_; ++t)
      for (int kind = 0; kind < 2 && cnt < P_; ++kind)
        for (int j = 0; j < SM_ && cnt < P_; ++j) {
          int v = (kind == 0) ? (amax[j] == 4 + 4 * t) : (int)accB[t][j];
          if (v) {
            int p = (kind == 0) ? j : nextE[t][j];
            int e = (kind == 0) ? (j + 1) : p;
            int slot = b * P_ + cnt;
            j0_ws[slot] = j; e_ws[slot] = e;
            valid_ws[slot] = 1; pos_ws[slot] = p + 1;
            selpos[cnt] = p + 1;
            ++cnt;
          }
        }
    for (; cnt < P_; ++cnt) {
      int slot = b * P_ + cnt;
      j0_ws[slot] = 0; e_ws[slot] = 1; valid_ws[slot] = 0; pos_ws[slot] = -1;
      selpos[cnt] = -1;
    }
  }
  __syncthreads();
  if (tid < 90) {
    int i = tid / 9, r = tid % 9;
    int j = r + (r >= i ? 1 : 0);
    out_pp[((size_t)b * 90 + tid) * 2 + 0] = (float)selpos[i];
    out_pp[((size_t)b * 90 + tid) * 2 + 1] = (float)selpos[j];
  }
}

// ---------------------------------------------------------------------------
// K3: ents[b,p,:] = mean over span of [h[b,j+1,:], softmax(ner[b,j+1,:])],
// zero if invalid. One block per (b,p); each thread owns 4-5 dims of Dx=1041.
// ---------------------------------------------------------------------------
__global__ __launch_bounds__(256) void k3_ents(
    const float* __restrict__ hbuf, const float* __restrict__ ner_ws,
    const int* __restrict__ j0_ws, const int* __restrict__ e_ws,
    const int* __restrict__ valid_ws, float* __restrict__ ents)
{
  int slot = blockIdx.x;                 // b*P_ + p
  int b = slot / P_;
  int tid = threadIdx.x;
  float* outp = ents + (size_t)slot * DX_;
  if (!valid_ws[slot]) {
    for (int i = 0; i < 5; ++i) {
      int d = tid + 256 * i;
      if (d < DX_) outp[d] = 0.f;
    }
    return;
  }
  int j0 = j0_ws[slot], e = e_ws[slot];
  float acc[5] = {0.f, 0.f, 0.f, 0.f, 0.f};
  for (int j = j0; j < e; ++j) {
    int srow = b * S_ + j + 1;
    for (int i = 0; i < 5; ++i) {
      int d = tid + 256 * i;
      if (d < D_) {
        acc[i] += hbuf[(size_t)srow * D_ + d];
      } else if (d < DX_) {
        const float* nr = ner_ws + (size_t)srow * NER_;
        float mx = nr[0];
        for (int o = 1; o < NER_; ++o) mx = fmaxf(mx, nr[o]);
        float ssum = 0.f;
        for (int o = 0; o < NER_; ++o) ssum += expf(nr[o] - mx);
        acc[i] += expf(nr[d - D_] - mx) / ssum;
      }
    }
  }
  int len = e - j0; if (len < 1) len = 1;
  float inv = 1.f / (float)len;
  for (int i = 0; i < 5; ++i) {
    int d = tid + 256 * i;
    if (d < DX_) outp[d] = acc[i] * inv;
  }
}

// ---------------------------------------------------------------------------
// Generic fp32 WMMA GEMM: C[m,n] = act(sum_k A[m,k]*W(k,n) + bias[n]).
// W element address = (n/BLK)*S0 + k*S1 + (n%BLK):
//   - row-major (K x N): BLK=N, S0=0, S1=N
//   - Wbil(o,h,k) as (512 x 8192) with n=o*512+k: BLK=512, S0=512*512, S1=512
// 64x64 block tile, zero-padded K/M/N edges, 8 waves each 16x32 of output.
// ---------------------------------------------------------------------------
#define G_BM 64
#define G_BN 64
#define G_KB 32

__global__ __launch_bounds__(256) void gemm_wmma(
    const float* __restrict__ A, const float* __restrict__ W,
    const float* __restrict__ bias, float* __restrict__ C,
    int M, int N, int K, long BLK, long S0, long S1, int do_relu, int has_bias)
{
  __shared__ __align__(16) float As[G_BM * 34];   // stride 34
  __shared__ __align__(16) float Bs2[16 * 160];   // pair-interleaved
  const int tid   = threadIdx.x;
  const int wave  = tid >> 5;
  const int lane  = tid & 31;
  const int lhalf = lane >> 4;
  const int l16   = lane & 15;
  const int wm = wave >> 1, wn = wave & 1;
  const int m0 = blockIdx.y * G_BM, n0 = blockIdx.x * G_BN;

  v8f cacc[2];
  for (int t = 0; t < 2; ++t)
    for (int r = 0; r < 8; ++r) cacc[t][r] = 0.f;

  const int nstages = (K + G_KB - 1) / G_KB;
  for (int ks = 0; ks < nstages; ++ks) {
    int k0 = ks * G_KB;
    __syncthreads();
    for (int it = 0; it < 8; ++it) {
      int s = tid + 256 * it;                  // 2048 elements
      int row = s >> 5, cc = s & 31;
      int gm = m0 + row, gk = k0 + cc;
      As[row * 34 + cc] = (gm < M && gk < K) ? A[(size_t)gm * K + gk] : 0.f;
    }
    for (int it = 0; it < 8; ++it) {
      int s = tid + 256 * it;                  // 2048 elements
      int row = s >> 6, cc = s & 63;
      int gk = k0 + row, gn = n0 + cc;
      float v = 0.f;
      if (gk < K && gn < N) {
        size_t addr = (size_t)((long)gn / BLK) * (size_t)S0 +
                      (size_t)gk * (size_t)S1 + (size_t)((long)gn % BLK);
        v = W[addr];
      }
      Bs2[(row >> 1) * 160 + cc * 2 + (row & 1)] = v;
    }
    __syncthreads();
    const float* arow  = As + (16 * wm + l16) * 34 + 2 * lhalf;
    const float* bbase = Bs2 + lhalf * 160 + wn * 64 + l16 * 2;
    for (int k = 0; k < G_KB; k += 4) {
      v2f a = *(const v2f*)(arow + k);
      const float* bk = bbase + (k >> 1) * 160;
      v2f bt0 = *(const v2f*)(bk);
      v2f bt1 = *(const v2f*)(bk + 32);
      cacc[0] = wmma_f32(a, bt0, cacc[0]);
      cacc[1] = wmma_f32(a, bt1, cacc[1]);
    }
  }
  for (int t = 0; t < 2; ++t) {
    int col = n0 + wn * 32 + t * 16 + l16;
    if (col >= N) continue;
    float bb = has_bias ? bias[col] : 0.f;
    for (int r = 0; r < 8; ++r) {
      int row = m0 + 16 * wm + r + 8 * lhalf;
      if (row >= M) continue;
      float v = cacc[t][r] + bb;
      if (do_relu && v < 0.f) v = 0.f;
      C[(size_t)row * N + col] = v;
    }
  }
}

// ---------------------------------------------------------------------------
// K4: re[b,q,o] = sum_k V[b,i,o,k]*tail[k] + sum_h head[h]*Wlin[o,h]
//               + sum_k tail[k]*Wlin[o,512+k] + bbil[o]
// One block per (b,q) pair; 16 partials per output, deterministic reduction.
// ---------------------------------------------------------------------------
__global__ __launch_bounds__(256) void k4_re(
    const float* __restrict__ headb, const float* __restrict__ tailb,
    const float* __restrict__ Vb, const float* __restrict__ Wlin,
    const float* __restrict__ bbil, float* __restrict__ out_re)
{
  __shared__ float hd[H_];
  __shared__ float tl[H_];
  __shared__ float red[RE_][17];
  int b = blockIdx.x / 90, q = blockIdx.x % 90;
  int i = q / 9, r = q % 9;
  int j = r + (r >= i ? 1 : 0);
  int tid = threadIdx.x;
  const float* hp = headb + (size_t)(b * P_ + i) * H_;
  const float* tp = tailb + (size_t)(b * P_ + j) * H_;
  for (int s = tid; s < H_; s += 256) { hd[s] = hp[s]; tl[s] = tp[s]; }
  __syncthreads();
  int o = tid >> 4, part = tid & 15;
  const float* vrow = Vb + (size_t)(b * P_ + i) * (RE_ * H_) + (size_t)o * H_;
  const float* wl = Wlin + (size_t)o * (2 * H_);
  float acc = 0.f;
  for (int k = part * 32; k < part * 32 + 32; ++k)
    acc += vrow[k] * tl[k] + hd[k] * wl[k] + tl[k] * wl[H_ + k];
  red[o][part] = acc;
  __syncthreads();
  if (part == 0) {
    float s = 0.f;
    for (int x = 0; x < 16; ++x) s += red[o][x];
    out_re[((size_t)b * 90 + q) * RE_ + o] = s + bbil[o];
  }
}

// ---------------------------------------------------------------------------
extern "C" void kernel_launch(void* const* d_in, const int* in_sizes, int n_in,
                              void* d_out, int out_size, void* d_ws, size_t ws_size,
                              hipStream_t stream)
{
  const float* h    = (const float*)d_in[0];
  const float* W1   = (const float*)d_in[1];
  const float* b1   = (const float*)d_in[2];
  const float* W2   = (const float*)d_in[3];
  const float* b2   = (const float*)d_in[4];
  const float* Wh1  = (const float*)d_in[5];
  const float* bh1  = (const float*)d_in[6];
  const float* Wh2  = (const float*)d_in[7];
  const float* bh2  = (const float*)d_in[8];
  const float* Wt1  = (const float*)d_in[9];
  const float* bt1  = (const float*)d_in[10];
  const float* Wt2  = (const float*)d_in[11];
  const float* bt2  = (const float*)d_in[12];
  const float* Wbil = (const float*)d_in[13];
  const float* bbil = (const float*)d_in[14];
  const float* Wlin = (const float*)d_in[15];

  char* ws = (char*)d_ws;
  size_t off = 0;
  auto take = [&](size_t nbytes) -> char* {
    char* p = ws + off;
    off += (nbytes + 255) & ~(size_t)255;
    return p;
  };
  float* ner_ws   = (float*)take((size_t)BN_ * S_ * NER_ * 4);     // 2.23 MB
  int*   j0_ws    = (int*)take(BN_ * P_ * 4);
  int*   e_ws     = (int*)take(BN_ * P_ * 4);
  int*   valid_ws = (int*)take(BN_ * P_ * 4);
  int*   pos_ws   = (int*)take(BN_ * P_ * 4);
  float* ents     = (float*)take((size_t)BN_ * P_ * DX_ * 4);      // 1.33 MB
  float* t1       = (float*)take((size_t)BN_ * P_ * H_ * 4);
  float* t2       = (float*)take((size_t)BN_ * P_ * H_ * 4);
  float* headb    = (float*)take((size_t)BN_ * P_ * H_ * 4);
  float* tailb    = (float*)take((size_t)BN_ * P_ * H_ * 4);
  float* Vb       = (float*)take((size_t)BN_ * P_ * RE_ * H_ * 4); // 10.5 MB

  float* out     = (float*)d_out;
  float* out_ner = out;                                  // 32*1023*17
  float* out_pp  = out + (size_t)BN_ * SM_ * NER_;       // 32*90*2
  float* out_re  = out_pp + (size_t)BN_ * 90 * 2;        // 32*90*16

  // 1) ner = relu(h@W1+b1)@W2+b2   (fused, WMMA)
  k1_ner<<<dim3(256), dim3(256), 0, stream>>>(h, W1, b1, W2, b2, ner_ws, out_ner);
  // 2) argmax / span selection / pos_pairs
  k2_spans<<<dim3(BN_), dim3(256), 0, stream>>>(ner_ws, j0_ws, e_ws, valid_ws,
                                                pos_ws, out_pp);
  // 3) ents (segment means over [h, softmax(ner)])
  k3_ents<<<dim3(BN_ * P_), dim3(256), 0, stream>>>(h, ner_ws, j0_ws, e_ws,
                                                    valid_ws, ents);
  // 4) head/tail MLPs (WMMA)
  gemm_wmma<<<dim3(8, 5), dim3(256), 0, stream>>>(ents, Wh1, bh1, t1,
      320, 512, 1041, 512L, 0L, 512L, 1, 1);
  gemm_wmma<<<dim3(8, 5), dim3(256), 0, stream>>>(t1, Wh2, bh2, headb,
      320, 512, 512, 512L, 0L, 512L, 0, 1);
  gemm_wmma<<<dim3(8, 5), dim3(256), 0, stream>>>(ents, Wt1, bt1, t2,
      320, 512, 1041, 512L, 0L, 512L, 1, 1);
  gemm_wmma<<<dim3(8, 5), dim3(256), 0, stream>>>(t2, Wt2, bt2, tailb,
      320, 512, 512, 512L, 0L, 512L, 0, 1);
  // 5) bilinear transform V[b,i,o,k] = sum_h head * Wbil  (WMMA, strided W)
  gemm_wmma<<<dim3(128, 5), dim3(256), 0, stream>>>(headb, Wbil, nullptr, Vb,
      320, 8192, 512, 512L, (long)512 * 512, 512L, 0, 0);
  // 6) pairwise re
  k4_re<<<dim3(BN_ * 90), dim3(256), 0, stream>>>(headb, tailb, Vb, Wlin, bbil,
                                                  out_re);

  (void)in_sizes; (void)n_in; (void)out_size; (void)ws_size;
}